// BioSimulator_90829968376165
// MI455X (gfx1250) — compile-verified
//
#include <hip/hip_runtime.h>
#include <math.h>

typedef float v2f __attribute__((ext_vector_type(2)));
typedef float v8f __attribute__((ext_vector_type(8)));

#define NELEC   1024
#define OUTDIM  256
#define NPIX    (OUTDIM * OUTDIM)

// ---------------------------------------------------------------------------
// Kernel 1: per-electrode setup (wedge-dipole map + amplitude folding).
// Writes one float4 "A-row" per electrode:
//   [ 2*beta*vx, 2*beta*vy, -beta, -beta*(vx^2+vy^2) + log2(Bamp) ]
// so that exponent2[n,p] = A[n,:] . [px, py, px^2+py^2, 1]  (exp2 domain).
// ---------------------------------------------------------------------------
__global__ __launch_bounds__(256) void bio_setup(
    const float* __restrict__ stim,      // 1024 floats
    const float* __restrict__ pp,        // 13 floats
    float4* __restrict__ afeat)          // 1024 float4 (workspace)
{
  int n = blockIdx.x * 256 + threadIdx.x;
  if (n >= NELEC) return;

  const float kk = 17.3f, aa = 0.75f, bb = 120.0f;
  const float slope = 19152642.5f, half_ = 1.057e-07f, rheo = 2.39e-05f;
  const float freq = 300.0f, pw = 0.00017f, r2s = 0.5f;
  const float deg2pix = (float)OUTDIM / (2.0f * 30.0f);   // 4.266667

  // electrode grid: meshgrid(xc, xc, indexing="xy") => gx = xc[n%32], gy = xc[n/32]
  int col = n & 31, row = n >> 5;
  float gx = -15.0f + (float)col * (30.0f / 31.0f);
  float gy = -15.0f + (float)row * (30.0f / 31.0f);

  float dx = pp[10] * (1.0f / 300.0f);
  float dy = pp[11] * (1.0f / 300.0f);
  float th = pp[12] * (3.14159265358979323846f / 180.0f);
  float ct = __cosf(th), st = __sinf(th);

  float gxn = gx * ct - gy * st + dx;
  float gyn = gx * st + gy * ct + dy;

  // e = exp((gxn + i*gyn)/k)
  float wr = gxn * (1.0f / kk), wi = gyn * (1.0f / kk);
  float em = __expf(wr);
  float er = em * __cosf(wi), ei = em * __sinf(wi);

  // z = a*b*(e-1)/(b - a*e)  (complex division)
  float ab = aa * bb;
  float nr = ab * (er - 1.0f), ni = ab * ei;
  float dr = bb - aa * er,     di = -aa * ei;
  float inv = 1.0f / (dr * dr + di * di);
  float zr = (nr * dr + ni * di) * inv;
  float zi = (ni * dr - nr * di) * inv;

  float r2 = zr * zr + zi * zi;
  float r  = sqrtf(r2);
  float Mv = kk * (1.0f / (r + aa) - 1.0f / (r + bb));

  // stimulation -> amplitude + size
  float I    = stim[n] * 8e-05f;
  float Ieff = fmaxf(I - rheo, 0.0f);
  float Q    = Ieff * pw * freq;
  float Bamp = 1.0f / (1.0f + __expf(-slope * (Q - half_)));

  float rho       = pp[0];
  float size_base = sqrtf(I / (rho + 1e-09f));
  float sigma     = fmaxf(size_base * (r2s / (Mv + 1e-09f)) * deg2pix, 0.5f);

  // exponent in exp2 domain: beta = log2(e) * deg2pix^2 / (2*sigma^2)
  float alpha = (deg2pix * deg2pix) / (2.0f * sigma * sigma);
  float beta  = alpha * 1.4426950408889634f;
  float f3    = -beta * r2 + __log2f(fmaxf(Bamp, 1e-38f));

  afeat[n] = make_float4(2.0f * beta * zr, 2.0f * beta * zi, -beta, f3);
}

// ---------------------------------------------------------------------------
// Kernel 2: main accumulation. One wave32 per 16-pixel tile.
// Per 16-electrode group: ds_load_b64 A fragment, v_wmma_f32_16x16x4_f32,
// 8x v_exp_f32 + add. Then cross-half reduce, polynomial, clamp, store.
// ---------------------------------------------------------------------------
__global__ __launch_bounds__(256) void bio_main(
    const float* __restrict__ pp,        // 13 floats
    const float4* __restrict__ afeat,    // 1024 float4 from bio_setup
    float* __restrict__ outp)            // 65536 floats
{
  __shared__ float4 sA[NELEC];           // 16 KB of the 320 KB LDS

  int tid = threadIdx.x;
  // cooperative stage of the A-table into LDS (reused by all 8 waves)
  for (int i = tid; i < NELEC; i += 256) sA[i] = afeat[i];
  __syncthreads();

  int lane  = tid & 31;
  int wave  = tid >> 5;
  int gwave = blockIdx.x * 8 + wave;
  int pbase = gwave * 16;                // 16-pixel tile (never crosses a row: 256 % 16 == 0)
  int lo    = lane & 15;
  bool hi   = (lane >= 16);

  int p    = pbase + lo;
  int pcol = p & (OUTDIM - 1);
  int prow = p >> 8;
  const float step = 60.0f / 255.0f;
  float px = -30.0f + (float)pcol * step;
  float py = -30.0f + (float)prow * step;

  // B-matrix 4x16 fragment: VGPR v, lanes 0-15 -> K=v, lanes 16-31 -> K=v+2
  // rows: [px, py, px^2+py^2, 1]
  v2f bfrag;
  bfrag.x = hi ? (px * px + py * py) : px;
  bfrag.y = hi ? 1.0f : py;

  const v2f* sA2 = (const v2f*)sA;       // 8-byte aligned pairs
  float acc = 0.0f;

  #pragma unroll 4
  for (int m = 0; m < NELEC; m += 16) {
    // A-matrix 16x4 fragment: lane holds row M = lane%16,
    // lanes 0-15 -> K={0,1}, lanes 16-31 -> K={2,3}
    int idx2 = (m + lo) * 2 + (hi ? 1 : 0);
    v2f af = sA2[idx2];                  // ds_load_b64

    v8f c = {};
    v8f d = __builtin_amdgcn_wmma_f32_16x16x4_f32(
        /*neg_a=*/false, af, /*neg_b=*/false, bfrag,
        /*c_mod=*/(short)0, c, /*reuse_a=*/false, /*reuse_b=*/false);

    // D[vgpr j] = exponent2 for electrode (m + j + 8*hi) at pixel p.
    // out += sum_n Bamp_n * 2^exponent2  (Bamp folded into the constant term)
    float e = 0.0f;
    e += __builtin_amdgcn_exp2f(d[0]);
    e += __builtin_amdgcn_exp2f(d[1]);
    e += __builtin_amdgcn_exp2f(d[2]);
    e += __builtin_amdgcn_exp2f(d[3]);
    e += __builtin_amdgcn_exp2f(d[4]);
    e += __builtin_amdgcn_exp2f(d[5]);
    e += __builtin_amdgcn_exp2f(d[6]);
    e += __builtin_amdgcn_exp2f(d[7]);
    acc += e;
  }

  // pixel p lives in lane p%16 and lane p%16+16 -> fold the two halves
  acc += __shfl_xor(acc, 16, 32);

  if (lane < 16) {
    float a0 = pp[3], a1 = pp[4], a2 = pp[5], a3 = pp[6], a4 = pp[7];
    float o = acc;
    float poly = a0 + o * (a1 + o * (a2 + o * (a3 + o * a4)));
    outp[p] = fminf(fmaxf(poly, 0.0f), 1.0f);
  }
}

// ---------------------------------------------------------------------------
extern "C" void kernel_launch(void* const* d_in, const int* in_sizes, int n_in,
                              void* d_out, int out_size, void* d_ws, size_t ws_size,
                              hipStream_t stream) {
  const float* stim = (const float*)d_in[0];   // (1,1,32,32) fp32
  const float* pp   = (const float*)d_in[1];   // (1,13) fp32
  float* outp       = (float*)d_out;           // (1,1,256,256) fp32
  float4* afeat     = (float4*)d_ws;           // 16 KB scratch

  bio_setup<<<NELEC / 256, 256, 0, stream>>>(stim, pp, afeat);

  // 65536 pixels / 16 per wave = 4096 waves; 8 waves (256 threads) per block
  bio_main<<<(NPIX / 16) / 8, 256, 0, stream>>>(pp, afeat, outp);
}